// LRMCModule_73297911873905
// MI455X (gfx1250) — compile-verified
//
#include <hip/hip_runtime.h>

#define Dd 4096
#define Rr 128
#define JS 8          // split of the streamed dimension (grid.y)
#define JCOLS (Dd / JS)   // 512 columns (or rows) per split block

typedef float v2f __attribute__((ext_vector_type(2)));
typedef float v8f __attribute__((ext_vector_type(8)));

__device__ __forceinline__ v8f wmma4(v2f a, v2f b, v8f c) {
  // D = A(16x4,f32) * B(4x16,f32) + C(16x16,f32)
  return __builtin_amdgcn_wmma_f32_16x16x4_f32(
      false, a, false, b, (short)0, c, false, false);
}

// ---- CDNA5 async global->LDS copies (ASYNCcnt tracked) ----------------------
__device__ __forceinline__ void async_g2l_b32(const float* g, float* l) {
  unsigned off = (unsigned)(size_t)l;  // low 32 bits of generic addr = LDS offset
  asm volatile("global_load_async_to_lds_b32 %0, %1, off"
               :: "v"(off), "v"(g) : "memory");
}
__device__ __forceinline__ void async_g2l_b128(const float* g, float* l) {
  unsigned off = (unsigned)(size_t)l;
  asm volatile("global_load_async_to_lds_b128 %0, %1, off"
               :: "v"(off), "v"(g) : "memory");
}
__device__ __forceinline__ void wait_async0() {
  asm volatile("s_wait_asynccnt 0x0" ::: "memory");
}

// -----------------------------------------------------------------------------
// Kernel 1: grid (64, 8), 256 threads. Block (bx,js): rows [64bx,64bx+64),
// cols [512js, 512js+512).
//   error = mask*(x - Uold@V)  -> global err
//   dUpart[js] = error @ V^T   (partial, summed later)
// -----------------------------------------------------------------------------
__global__ __launch_bounds__(256) void err_du_kernel(
    const float* __restrict__ x, const float* __restrict__ mask,
    const float* __restrict__ V, const float* __restrict__ U,
    float* __restrict__ err, float* __restrict__ dUp)
{
  __shared__ __attribute__((aligned(16))) float Up[64][Rr + 4];      // U panel
  __shared__ __attribute__((aligned(16))) float Vt[2][32][Rr + 6];   // Vt[b][c][r]
  __shared__ __attribute__((aligned(16))) float Et[64][32 + 4];      // error tile

  const int tid  = threadIdx.x;
  const int lane = tid & 31;
  const int wave = tid >> 5;
  const int lo   = lane & 15;
  const int hi   = lane >> 4;
  const int rm   = (wave & 3) * 16;
  const int ch   = wave >> 2;        // 0..1 column half
  const int cn   = ch * 16;
  const int m0   = blockIdx.x * 64;
  const int js   = blockIdx.y;
  const int jb   = js * JCOLS;
  const int JITER = JCOLS / 32;      // 16

  // async load of U panel (64 x 128), b128 contiguous both sides
  for (int idx = tid; idx < 64 * (Rr / 4); idx += 256) {
    int r = idx >> 5, c4 = idx & 31;
    async_g2l_b128(U + (size_t)(m0 + r) * Rr + c4 * 4, &Up[r][c4 * 4]);
  }
  // prefetch first V tile (transposed per-element)
  for (int idx = tid; idx < Rr * 32; idx += 256) {
    int r = idx >> 5, c = idx & 31;
    async_g2l_b32(V + (size_t)r * Dd + (jb + c), &Vt[0][c][r]);
  }

  v8f du[4] = {};

  for (int jj = 0; jj < JITER; ++jj) {
    const int cur = jj & 1;
    wait_async0();        // this wave's async copies landed
    __syncthreads();      // everyone's landed; prev readers of buffers done

    if (jj + 1 < JITER) {   // prefetch next V tile into other buffer
      const int jn = jb + (jj + 1) * 32;
      for (int idx = tid; idx < Rr * 32; idx += 256) {
        int r = idx >> 5, c = idx & 31;
        async_g2l_b32(V + (size_t)r * Dd + (jn + c), &Vt[cur ^ 1][c][r]);
      }
    }

    // P = U_panel @ V_tile  (one 16x16 tile per wave), K = 128
    v8f p = {};
#pragma unroll
    for (int k = 0; k < Rr; k += 4) {
      v2f a = *(const v2f*)&Up[rm + lo][k + 2 * hi];
      v2f b = *(const v2f*)&Vt[cur][cn + lo][k + 2 * hi];
      p = wmma4(a, b, p);
    }

    // error tile: e = mask*(x - P); -> global err and LDS Et
    const int colg = jb + jj * 32 + cn + lo;
#pragma unroll
    for (int i = 0; i < 8; ++i) {
      int rl = rm + i + 8 * hi;
      size_t g = (size_t)(m0 + rl) * Dd + colg;
      float e = mask[g] * (x[g] - p[i]);
      err[g] = e;
      Et[rl][cn + lo] = e;
    }
    __syncthreads();

    // dU += err_tile(64x32) @ V_tile^T(32x128); wave: rows rm, cols [64ch,64ch+64)
#pragma unroll
    for (int kk = 0; kk < 32; kk += 4) {
      v2f a = *(const v2f*)&Et[rm + lo][kk + 2 * hi];
#pragma unroll
      for (int t = 0; t < 4; ++t) {
        int nb = ch * 64 + t * 16;
        v2f b;
        b.x = Vt[cur][kk + 2 * hi][nb + lo];
        b.y = Vt[cur][kk + 2 * hi + 1][nb + lo];
        du[t] = wmma4(a, b, du[t]);
      }
    }
  }

  // write partial dU (exclusively owned slice of dUp[js])
  float* dst = dUp + (size_t)js * Dd * Rr;
#pragma unroll
  for (int t = 0; t < 4; ++t) {
    int cb = ch * 64 + t * 16 + lo;
#pragma unroll
    for (int i = 0; i < 8; ++i) {
      int rl = rm + i + 8 * hi;
      dst[(size_t)(m0 + rl) * Rr + cb] = du[t][i];
    }
  }
}

// -----------------------------------------------------------------------------
// Kernel 2: grid (64, 8), 256 threads. Block (bx,is): V cols [64bx,64bx+64),
// err rows [512is, 512is+512).  dVpart[is] = Unew^T @ err  (partial)
// -----------------------------------------------------------------------------
__global__ __launch_bounds__(256) void dv_kernel(
    const float* __restrict__ U, const float* __restrict__ err,
    float* __restrict__ dVp)
{
  __shared__ __attribute__((aligned(16))) float UcT[2][Rr][32 + 4]; // UcT[b][m][k]
  __shared__ __attribute__((aligned(16))) float EcT[2][64][32 + 4]; // EcT[b][n][k]

  const int tid = threadIdx.x, lane = tid & 31, wave = tid >> 5;
  const int lo = lane & 15, hi = lane >> 4;
  const int mt = wave * 16;
  const int j0 = blockIdx.x * 64;
  const int ib = blockIdx.y * JCOLS;
  const int CHUNKS = JCOLS / 32;     // 16

  // prefetch chunk 0 (both tiles, transposed per-element)
  for (int idx = tid; idx < 32 * Rr; idx += 256) {
    int r = idx >> 7, m = idx & 127;
    async_g2l_b32(U + (size_t)(ib + r) * Rr + m, &UcT[0][m][r]);
  }
  for (int idx = tid; idx < 32 * 64; idx += 256) {
    int r = idx >> 6, c = idx & 63;
    async_g2l_b32(err + (size_t)(ib + r) * Dd + (j0 + c), &EcT[0][c][r]);
  }

  v8f acc[4] = {};
  for (int cc = 0; cc < CHUNKS; ++cc) {
    const int cur = cc & 1;
    wait_async0();
    __syncthreads();

    if (cc + 1 < CHUNKS) {
      const int in = ib + (cc + 1) * 32;
      for (int idx = tid; idx < 32 * Rr; idx += 256) {
        int r = idx >> 7, m = idx & 127;
        async_g2l_b32(U + (size_t)(in + r) * Rr + m, &UcT[cur ^ 1][m][r]);
      }
      for (int idx = tid; idx < 32 * 64; idx += 256) {
        int r = idx >> 6, c = idx & 63;
        async_g2l_b32(err + (size_t)(in + r) * Dd + (j0 + c), &EcT[cur ^ 1][c][r]);
      }
    }

#pragma unroll
    for (int kk = 0; kk < 32; kk += 4) {
      v2f a = *(const v2f*)&UcT[cur][mt + lo][kk + 2 * hi];   // b64 contiguous
#pragma unroll
      for (int t = 0; t < 4; ++t) {
        v2f b = *(const v2f*)&EcT[cur][t * 16 + lo][kk + 2 * hi];
        acc[t] = wmma4(a, b, acc[t]);
      }
    }
  }

  float* dst = dVp + (size_t)blockIdx.y * Rr * Dd;
#pragma unroll
  for (int t = 0; t < 4; ++t)
#pragma unroll
    for (int i = 0; i < 8; ++i) {
      int r = mt + i + 8 * hi;
      dst[(size_t)r * Dd + j0 + t * 16 + lo] = acc[t][i];
    }
}

// -----------------------------------------------------------------------------
// Deterministic reduction: Out = Base + sum_{s<8} Part[s]   (524288 floats)
// grid = 512, 256 threads, float4 per thread
// -----------------------------------------------------------------------------
__global__ __launch_bounds__(256) void reduce_add_kernel(
    const float* __restrict__ base, const float* __restrict__ parts,
    float* __restrict__ out)
{
  size_t idx = (size_t)blockIdx.x * 256 + threadIdx.x;      // float4 index
  float4 s = ((const float4*)base)[idx];
#pragma unroll
  for (int p = 0; p < JS; ++p) {
    float4 d = ((const float4*)(parts + (size_t)p * Dd * Rr))[idx];
    s.x += d.x; s.y += d.y; s.z += d.z; s.w += d.w;
  }
  ((float4*)out)[idx] = s;
}

// -----------------------------------------------------------------------------
// Kernel 3: out = U @ V.  grid = (128 N-tiles of 32, 64 M-panels of 64)
// -----------------------------------------------------------------------------
__global__ __launch_bounds__(256) void final_kernel(
    const float* __restrict__ U, const float* __restrict__ V, float* __restrict__ out)
{
  __shared__ __attribute__((aligned(16))) float Up[64][Rr + 4];
  __shared__ __attribute__((aligned(16))) float Vt[32][Rr + 6];

  const int tid = threadIdx.x, lane = tid & 31, wave = tid >> 5;
  const int lo = lane & 15, hi = lane >> 4;
  const int rm = (wave & 3) * 16, cn = (wave >> 2) * 16;
  const int m0 = blockIdx.y * 64, j0 = blockIdx.x * 32;

  for (int idx = tid; idx < 64 * (Rr / 4); idx += 256) {
    int r = idx >> 5, c4 = idx & 31;
    async_g2l_b128(U + (size_t)(m0 + r) * Rr + c4 * 4, &Up[r][c4 * 4]);
  }
  for (int idx = tid; idx < Rr * 32; idx += 256) {
    int r = idx >> 5, c = idx & 31;
    async_g2l_b32(V + (size_t)r * Dd + (j0 + c), &Vt[c][r]);
  }
  wait_async0();
  __syncthreads();

  v8f p = {};
#pragma unroll
  for (int k = 0; k < Rr; k += 4) {
    v2f a = *(const v2f*)&Up[rm + lo][k + 2 * hi];
    v2f b = *(const v2f*)&Vt[cn + lo][k + 2 * hi];
    p = wmma4(a, b, p);
  }
#pragma unroll
  for (int i = 0; i < 8; ++i)
    out[(size_t)(m0 + rm + i + 8 * hi) * Dd + (j0 + cn + lo)] = p[i];
}

// -----------------------------------------------------------------------------
extern "C" void kernel_launch(void* const* d_in, const int* in_sizes, int n_in,
                              void* d_out, int out_size, void* d_ws, size_t ws_size,
                              hipStream_t stream) {
  const float* x    = (const float*)d_in[0];
  const float* mask = (const float*)d_in[1];
  const float* U0   = (const float*)d_in[2];
  const float* V0   = (const float*)d_in[3];
  float* out = (float*)d_out;

  const size_t DR = (size_t)Dd * Rr;      // 524288
  const size_t DD = (size_t)Dd * Dd;

  float* UA  = (float*)d_ws;              // U ping
  float* UB  = UA + DR;                   // U pong
  float* VV  = UB + DR;                   // V (updated in place via reduce)
  float* E   = VV + DR;                   // error, 64 MB
  float* dUp = E + DD;                    // 8 partial dU
  float* dVp = dUp + (size_t)JS * DR;     // 8 partial dV

  hipMemcpyAsync(UA, U0, DR * sizeof(float), hipMemcpyDeviceToDevice, stream);
  hipMemcpyAsync(VV, V0, DR * sizeof(float), hipMemcpyDeviceToDevice, stream);

  float* ucur = UA;
  float* unext = UB;
  for (int it = 0; it < 3; ++it) {
    err_du_kernel<<<dim3(64, JS), 256, 0, stream>>>(x, mask, VV, ucur, E, dUp);
    reduce_add_kernel<<<512, 256, 0, stream>>>(ucur, dUp, unext);   // Unew
    dv_kernel<<<dim3(64, JS), 256, 0, stream>>>(unext, E, dVp);
    reduce_add_kernel<<<512, 256, 0, stream>>>(VV, dVp, VV);        // Vnew
    float* t = ucur; ucur = unext; unext = t;
  }
  final_kernel<<<dim3(128, 64), 256, 0, stream>>>(ucur, VV, out);
}